// MambaBlock_58746562674795
// MI455X (gfx1250) — compile-verified
//
#include <hip/hip_runtime.h>
#include <hip/hip_bf16.h>

#define D_MODEL   1024
#define D_STATE   64
#define D_CONV    4
#define HEADDIM   64
#define D_INNER   2048
#define NHEADS    32
#define CONV_DIM  2176
#define D_IN_PROJ 4256
#define ZX_LD     4288      // D_IN_PROJ padded to a multiple of 64 (GEMM N-tile)
#define BATCH     2
#define SEQLEN    4096
#define NTOK      (BATCH * SEQLEN)
#define EPS_F     1e-5f

// GEMM tiling
#define BM 128
#define BN 64
#define KC 32
#define AS_STRIDE 34        // 32 + TDM pad (2 dwords per 32) -> bank-conflict-free A reads

typedef __attribute__((ext_vector_type(2))) float v2f;
typedef __attribute__((ext_vector_type(8))) float v8f;
typedef __attribute__((ext_vector_type(4))) unsigned int u32x4;
typedef __attribute__((ext_vector_type(8))) int i32x8;
typedef __attribute__((ext_vector_type(4))) int i32x4;

// ---------------------------------------------------------------- utilities
__device__ __forceinline__ float block_reduce_sum(float v, float* red) {
  int tid = threadIdx.x;
  red[tid] = v;
  __syncthreads();
  for (int s = 128; s > 0; s >>= 1) {
    if (tid < s) red[tid] += red[tid + s];
    __syncthreads();
  }
  float r = red[0];
  __syncthreads();
  return r;
}

// Flat addr of an LDS object: low 32 bits are the LDS byte offset (aperture doc 10.2)
__device__ __forceinline__ unsigned lds_off(const void* p) {
  return (unsigned)(unsigned long long)p;
}

// Issue one 2-D TDM tile load: global (tile_x x tile_y elems, row stride
// stride_elems, fp32) -> LDS at lds_byte.  g1w0 carries data_size/pad config.
__device__ __forceinline__ void tdm_load_2d(const void* gptr, unsigned lds_byte,
                                            int tile_x, int tile_y,
                                            long long stride_elems, unsigned g1w0) {
  unsigned long long ga = (unsigned long long)gptr;
  u32x4 g0;
  g0.x = 1u;                                            // count=1 (valid user D#)
  g0.y = lds_byte;                                      // lds_addr
  g0.z = (unsigned)(ga & 0xFFFFFFFFu);                  // global_addr[31:0]
  g0.w = (unsigned)((ga >> 32) & 0x01FFFFFFu) | (2u << 30);  // addr[56:32] | type=2
  unsigned td0 = (unsigned)tile_x;                      // tensor_dim = tile_dim (no clip)
  unsigned td1 = (unsigned)tile_y;
  unsigned long long s0 = (unsigned long long)stride_elems;
  i32x8 g1;
  g1[0] = (int)g1w0;                                    // mask=0,data_size=4B,pad cfg
  g1[1] = (int)((td0 & 0xFFFFu) << 16);                 // tensor_dim0[15:0]
  g1[2] = (int)((td0 >> 16) | ((td1 & 0xFFFFu) << 16)); // td0 hi | tensor_dim1 lo
  g1[3] = (int)((td1 >> 16) | ((unsigned)tile_x << 16));// td1 hi | tile_dim0
  g1[4] = (int)((unsigned)tile_y & 0xFFFFu);            // tile_dim1 (tile_dim2=0)
  g1[5] = (int)(s0 & 0xFFFFFFFFu);                      // tensor_dim0_stride lo
  g1[6] = (int)((s0 >> 32) & 0xFFFFu);                  // stride hi | dim1_stride=0
  g1[7] = 0;
  i32x4 z4 = {0, 0, 0, 0};
  i32x8 z8 = {0, 0, 0, 0, 0, 0, 0, 0};
  __builtin_amdgcn_tensor_load_to_lds(g0, g1, z4, z4, z8, 0);
}

#define G1_PLAIN 0x00020000u                              // data_size=4B
#define G1_PAD32 (0x00020000u | (1u<<20) | (4u<<22) | (1u<<25)) // +2 dwords per 32

// Wt[k*ldo + n] = (n < rows) ? W[n*cols + k] : 0   (zero-fills N padding)
__global__ void transpose_pad_kernel(const float* __restrict__ W, float* __restrict__ Wt,
                                     int rows, int cols, int ldo) {
  size_t i = (size_t)blockIdx.x * blockDim.x + threadIdx.x;
  size_t total = (size_t)cols * ldo;
  if (i >= total) return;
  int k = (int)(i / ldo);
  int n = (int)(i % ldo);
  Wt[i] = (n < rows) ? W[(size_t)n * cols + k] : 0.f;
}

// ---------------------------------------------------------------- layernorm
__global__ void ln_kernel(const float* __restrict__ x, const float* __restrict__ w,
                          const float* __restrict__ b, float* __restrict__ u) {
  __shared__ float red[256];
  size_t t = blockIdx.x;
  const float* xr = x + t * D_MODEL;
  float s = 0.f;
  for (int i = threadIdx.x; i < D_MODEL; i += 256) s += xr[i];
  float mu = block_reduce_sum(s, red) * (1.f / D_MODEL);
  float v = 0.f;
  for (int i = threadIdx.x; i < D_MODEL; i += 256) { float d = xr[i] - mu; v += d * d; }
  float var = block_reduce_sum(v, red) * (1.f / D_MODEL);
  float rstd = rsqrtf(var + EPS_F);
  float* ur = u + t * D_MODEL;
  for (int i = threadIdx.x; i < D_MODEL; i += 256)
    ur[i] = (xr[i] - mu) * rstd * w[i] + b[i];
}

// --------------------------------------------- WMMA GEMM with TDM double-buffer
// C[M,N] = A[M,K] * Bt[K,N] (+resid).  M%128==0, N%64==0, K%32==0.
// 8 waves; wave w owns rows 16w..16w+15 of the 128x64 block tile (4 WMMA chains).
__global__ void __launch_bounds__(256)
gemm_wmma_tdm_kernel(const float* __restrict__ A, const float* __restrict__ Bt,
                     float* __restrict__ C, const float* __restrict__ resid,
                     int M, int N, int K) {
  __shared__ float As[2][BM * AS_STRIDE];   // 2 x 17408 B
  __shared__ float Bs[2][KC * BN];          // 2 x  8192 B
  const int tid   = threadIdx.x;
  const int lane  = tid & 31;
  const int wave  = tid >> 5;
  const int lrow  = lane & 15;
  const int khalf = (lane >> 4) << 1;       // 0 (lanes 0-15) / 2 (lanes 16-31)
  const int m0 = blockIdx.y * BM;
  const int n0 = blockIdx.x * BN;

  v8f acc[4] = {};

  const int nch = K / KC;
  if (wave == 0) {   // wave-uniform: scalar branch, EXEC untouched elsewhere
    tdm_load_2d(A + (size_t)m0 * K, lds_off(&As[0][0]), KC, BM, K, G1_PAD32);
    tdm_load_2d(Bt + n0,            lds_off(&Bs[0][0]), BN, KC, N, G1_PLAIN);
    __builtin_amdgcn_s_wait_tensorcnt(0);
  }
  __syncthreads();

  for (int ch = 0; ch < nch; ++ch) {
    const int buf = ch & 1;
    if (wave == 0 && ch + 1 < nch) {        // prefetch next K-chunk
      const int k1 = (ch + 1) * KC;
      tdm_load_2d(A + (size_t)m0 * K + k1, lds_off(&As[buf ^ 1][0]), KC, BM, K, G1_PAD32);
      tdm_load_2d(Bt + (size_t)k1 * N + n0, lds_off(&Bs[buf ^ 1][0]), BN, KC, N, G1_PLAIN);
    }

    const float* as = &As[buf][(wave * 16 + lrow) * AS_STRIDE];
#pragma unroll
    for (int kk = 0; kk < KC; kk += 4) {
      float2 av = *(const float2*)(as + kk + khalf);   // A frag: 16x4 fp32 layout
      v2f a; a.x = av.x; a.y = av.y;
      const float* b0 = &Bs[buf][(kk + khalf) * BN];
      const float* b1 = b0 + BN;
#pragma unroll
      for (int t = 0; t < 4; ++t) {                    // 4 N-tiles: independent chains
        v2f bfrag; bfrag.x = b0[t * 16 + lrow]; bfrag.y = b1[t * 16 + lrow];
        acc[t] = __builtin_amdgcn_wmma_f32_16x16x4_f32(false, a, false, bfrag,
                                                       (short)0, acc[t], false, false);
      }
    }

    if (wave == 0) __builtin_amdgcn_s_wait_tensorcnt(0);
    __syncthreads();                        // publish next buffer / retire this one
  }

  const int rbase = m0 + wave * 16 + ((lane >> 4) << 3);  // lanes 16-31: M=8..15
#pragma unroll
  for (int t = 0; t < 4; ++t) {
    const int col = n0 + t * 16 + lrow;
#pragma unroll
    for (int r = 0; r < 8; ++r) {
      size_t o = (size_t)(rbase + r) * N + col;
      float v = acc[t][r];
      if (resid) v += resid[o];
      C[o] = v;
    }
  }
}

// ------------------------------------------------- depthwise causal conv + SiLU
__global__ void conv_silu_kernel(const float* __restrict__ zx, const float* __restrict__ cw,
                                 const float* __restrict__ cb, float* __restrict__ xc) {
  size_t i = (size_t)blockIdx.x * 256 + threadIdx.x;
  if (i >= (size_t)NTOK * CONV_DIM) return;
  int    c = (int)(i % CONV_DIM);
  size_t t = i / CONV_DIM;
  int    l = (int)(t & (SEQLEN - 1));
  size_t base = (t - (size_t)l) * ZX_LD + D_INNER + c;
  float acc = cb[c];
#pragma unroll
  for (int k = 0; k < D_CONV; k++) {
    int ls = l - (D_CONV - 1) + k;
    if (ls >= 0) acc += zx[base + (size_t)ls * ZX_LD] * cw[c * D_CONV + k];
  }
  float sig = 1.f / (1.f + expf(-acc));
  xc[i] = acc * sig;
}

// ---------------------------------------------------------------- dt softplus
__global__ void dt_softplus_kernel(const float* __restrict__ zx,
                                   const float* __restrict__ dt_bias,
                                   float* __restrict__ dts) {
  int i = blockIdx.x * 256 + threadIdx.x;
  if (i >= NTOK * NHEADS) return;
  int    h = i & (NHEADS - 1);
  size_t t = (size_t)(i >> 5);
  float  v = zx[t * ZX_LD + (D_INNER + CONV_DIM) + h] + dt_bias[h];
  dts[i] = (v > 20.f) ? v : log1pf(expf(v));
}

// ---------------------------------------------------------------- SSM scan
__global__ void scan_kernel(const float* __restrict__ xc, const float* __restrict__ dts,
                            const float* __restrict__ A_log, const float* __restrict__ Dp,
                            float* __restrict__ ys) {
  const int b   = blockIdx.x >> 5;
  const int h   = blockIdx.x & 31;
  const int tid = threadIdx.x;
  const int p   = tid >> 2;
  const int n0  = (tid & 3) << 4;
  const float Ah = -expf(A_log[h]);
  const float Dh = Dp[h];

  __shared__ float sx[HEADDIM];
  __shared__ float sB[D_STATE];
  __shared__ float sC[D_STATE];
  __shared__ float sdt[1];

  float hst[16];
#pragma unroll
  for (int j = 0; j < 16; j++) hst[j] = 0.f;

  for (int l = 0; l < SEQLEN; l++) {
    size_t tok = (size_t)b * SEQLEN + l;
    __syncthreads();
    if (tid < 64)        sx[tid]       = xc[tok * CONV_DIM + h * HEADDIM + tid];
    else if (tid < 128)  sB[tid - 64]  = xc[tok * CONV_DIM + D_INNER + (tid - 64)];
    else if (tid < 192)  sC[tid - 128] = xc[tok * CONV_DIM + D_INNER + D_STATE + (tid - 128)];
    else if (tid == 192) sdt[0]        = dts[tok * NHEADS + h];
    __syncthreads();

    float dt  = sdt[0];
    float dA  = expf(dt * Ah);
    float dtx = dt * sx[p];
    float part = 0.f;
#pragma unroll
    for (int j = 0; j < 16; j++) {
      hst[j] = hst[j] * dA + dtx * sB[n0 + j];
      part  += hst[j] * sC[n0 + j];
    }
    part += __shfl_xor(part, 1, 32);
    part += __shfl_xor(part, 2, 32);
    if ((tid & 3) == 0)
      ys[tok * D_INNER + h * HEADDIM + p] = part + Dh * sx[p];
  }
}

// ---------------------------------------------------------------- gate + RMSNorm
__global__ void gate_rms_kernel(const float* __restrict__ ys, const float* __restrict__ zx,
                                const float* __restrict__ nw, float* __restrict__ yn) {
  __shared__ float red[256];
  __shared__ float gv[D_INNER];
  size_t t = blockIdx.x;
  float ss = 0.f;
  for (int i = threadIdx.x; i < D_INNER; i += 256) {
    float z = zx[t * ZX_LD + i];
    float g = ys[t * D_INNER + i] * (z / (1.f + expf(-z)));
    gv[i] = g;
    ss += g * g;
  }
  float ms = block_reduce_sum(ss, red) * (1.f / D_INNER);
  float sc = rsqrtf(ms + EPS_F);
  for (int i = threadIdx.x; i < D_INNER; i += 256)
    yn[t * D_INNER + i] = gv[i] * sc * nw[i];
}

// ---------------------------------------------------------------- launch
extern "C" void kernel_launch(void* const* d_in, const int* in_sizes, int n_in,
                              void* d_out, int out_size, void* d_ws, size_t ws_size,
                              hipStream_t stream) {
  (void)in_sizes; (void)n_in; (void)out_size; (void)ws_size;
  const float* x       = (const float*)d_in[0];
  const float* ln_w    = (const float*)d_in[1];
  const float* ln_b    = (const float*)d_in[2];
  const float* W_in    = (const float*)d_in[3];
  const float* conv_w  = (const float*)d_in[4];
  const float* conv_b  = (const float*)d_in[5];
  const float* dt_bias = (const float*)d_in[6];
  const float* A_log   = (const float*)d_in[7];
  const float* Dp      = (const float*)d_in[8];
  const float* norm_w  = (const float*)d_in[9];
  const float* W_out   = (const float*)d_in[10];
  float* out = (float*)d_out;

  float* ws = (float*)d_ws;
  size_t off = 0;
  float* u     = ws + off; off += (size_t)NTOK * D_MODEL;      // 8192x1024
  float* zx    = ws + off; off += (size_t)NTOK * ZX_LD;        // 8192x4288
  float* xc    = ws + off; off += (size_t)NTOK * CONV_DIM;     // 8192x2176
  float* dts   = ws + off; off += (size_t)NTOK * NHEADS;       // 8192x32
  float* ys    = ws + off; off += (size_t)NTOK * D_INNER;      // 8192x2048
  float* yn    = ws + off; off += (size_t)NTOK * D_INNER;      // 8192x2048
  float* WtIn  = ws + off; off += (size_t)D_MODEL * ZX_LD;     // 1024x4288
  float* WtOut = ws + off; off += (size_t)D_INNER * D_MODEL;   // 2048x1024

  // 0) transpose weights (zero-filling N padding) so B tiles are row-contiguous
  {
    size_t total = (size_t)D_MODEL * ZX_LD;
    transpose_pad_kernel<<<(unsigned)((total + 255) / 256), 256, 0, stream>>>(
        W_in, WtIn, D_IN_PROJ, D_MODEL, ZX_LD);
  }
  {
    size_t total = (size_t)D_INNER * D_MODEL;
    transpose_pad_kernel<<<(unsigned)((total + 255) / 256), 256, 0, stream>>>(
        W_out, WtOut, D_MODEL, D_INNER, D_MODEL);
  }

  // 1) layernorm
  ln_kernel<<<NTOK, 256, 0, stream>>>(x, ln_w, ln_b, u);

  // 2) in-projection: zx = u @ W_in^T   (M=8192, N=4288(pad), K=1024)
  {
    dim3 grid(ZX_LD / BN, NTOK / BM);        // (67, 64)
    gemm_wmma_tdm_kernel<<<grid, 256, 0, stream>>>(u, WtIn, zx, nullptr,
                                                   NTOK, ZX_LD, D_MODEL);
  }

  // 3) depthwise causal conv + SiLU, and dt softplus
  {
    size_t total = (size_t)NTOK * CONV_DIM;
    conv_silu_kernel<<<(unsigned)((total + 255) / 256), 256, 0, stream>>>(zx, conv_w, conv_b, xc);
  }
  dt_softplus_kernel<<<(NTOK * NHEADS) / 256, 256, 0, stream>>>(zx, dt_bias, dts);

  // 4) sequential SSM scan: one block per (batch, head)
  scan_kernel<<<BATCH * NHEADS, 256, 0, stream>>>(xc, dts, A_log, Dp, ys);

  // 5) gate with SiLU(z) + RMSNorm
  gate_rms_kernel<<<NTOK, 256, 0, stream>>>(ys, zx, norm_w, yn);

  // 6) out-projection + residual: out = x + yn @ W_out^T  (M=8192, N=1024, K=2048)
  {
    dim3 grid(D_MODEL / BN, NTOK / BM);      // (16, 64)
    gemm_wmma_tdm_kernel<<<grid, 256, 0, stream>>>(yn, WtOut, out, x,
                                                   NTOK, D_MODEL, D_INNER);
  }
}